// OFGLoss_11605001634475
// MI455X (gfx1250) — compile-verified
//
#include <hip/hip_runtime.h>
#include <math.h>

// Problem dims (fixed by reference)
#define Hh 160
#define Ww 192
#define Dd 160
#define NVOX 4915200L          // Hh*Ww*Dd
#define PAD 4                  // WIN=9 -> pad 4
#define WINSZ 729.0f           // 9^3
#define LRc  0.1f
#define B1c  0.9f
#define B2c  0.999f
#define EPSc 1e-8f

typedef float v2f __attribute__((ext_vector_type(2)));
typedef float v8f __attribute__((ext_vector_type(8)));

// ---------------- elementwise init ----------------
__global__ void init_state(const float* __restrict__ initf, float* __restrict__ flow,
                           float* __restrict__ m, float* __restrict__ v, float* __restrict__ vh) {
    long i = blockIdx.x * (long)blockDim.x + threadIdx.x;
    if (i >= 3 * NVOX) return;
    flow[i] = initf[i];
    m[i] = 0.f; v[i] = 0.f; vh[i] = 0.f;
}

// ---------------- pure running-sum window pass (axis W), coalesced lanes ----
__global__ void ws_line_w(const float* __restrict__ src, float* __restrict__ dst) {
    long c = blockIdx.y;
    const float* s = src + c * NVOX;
    float*       o = dst + c * NVOX;
    int l = blockIdx.x * blockDim.x + threadIdx.x;
    if (l >= Hh * Dd) return;
    int h = l / Dd, d = l - h * Dd;
    long base = (long)h * Ww * Dd + d;
    const long stride = Dd;
    float ssum = 0.f;
    for (int i = 0; i < Ww + PAD; ++i) {
        if (i + 8 < Ww) __builtin_prefetch(&s[base + (long)(i + 8) * stride], 0, 1);
        float vv = (i < Ww) ? s[base + (long)i * stride] : 0.f;
        ssum += vv;
        int j = i - PAD;
        if (j >= 0) o[base + (long)j * stride] = ssum;
        if (i >= 2 * PAD) ssum -= s[base + (long)(i - 2 * PAD) * stride];
    }
}

// ------- fused axis-H running sums (producers computed on the fly) ----------
// Lines along H (stride W*D); thread index over (w,d) -> contiguous d.

// [J, J^2] from y (loop-invariant J moments, run once)
__global__ void ws_h_pointJ(const float* __restrict__ y, float* __restrict__ out) {
    int l = blockIdx.x * blockDim.x + threadIdx.x;
    if (l >= Ww * Dd) return;
    const long stride = (long)Ww * Dd;
    float s0 = 0.f, s1 = 0.f;
    for (int i = 0; i < Hh + PAD; ++i) {
        float a = (i < Hh) ? y[l + (long)i * stride] : 0.f;
        s0 += a; s1 += a * a;
        int j = i - PAD;
        if (j >= 0) {
            long q = l + (long)j * stride;
            out[q] = s0; out[NVOX + q] = s1;
        }
        if (i >= 2 * PAD) {
            float a2 = y[l + (long)(i - 2 * PAD) * stride];
            s0 -= a2; s1 -= a2 * a2;
        }
    }
}

// [I, I^2, I*J] from (xw, y)
__global__ void ws_h_point3(const float* __restrict__ xw, const float* __restrict__ y,
                            float* __restrict__ out) {
    int l = blockIdx.x * blockDim.x + threadIdx.x;
    if (l >= Ww * Dd) return;
    const long stride = (long)Ww * Dd;
    float s0 = 0.f, s1 = 0.f, s2 = 0.f;
    for (int i = 0; i < Hh + PAD; ++i) {
        float a = 0.f, b = 0.f;
        if (i < Hh) {
            long q = l + (long)i * stride;
            __builtin_prefetch(&xw[q + 4 * stride], 0, 1);
            a = xw[q]; b = y[q];
        }
        s0 += a; s1 += a * a; s2 += a * b;
        int j = i - PAD;
        if (j >= 0) {
            long q = l + (long)j * stride;
            out[q] = s0; out[NVOX + q] = s1; out[2 * NVOX + q] = s2;
        }
        if (i >= 2 * PAD) {
            long q = l + (long)(i - 2 * PAD) * stride;
            float a2 = xw[q], b2 = y[q];
            s0 -= a2; s1 -= a2 * a2; s2 -= a2 * b2;
        }
    }
}

// NCC adjoint coefficient fields from window sums:
// A = cross/den, B = cross^2*Jvar/den^2; fields {A, B, A*uJ, B*uI}
__device__ __forceinline__ void ncc_fields(const float* __restrict__ S,
                                           const float* __restrict__ Js,
                                           const float* __restrict__ J2s, long q,
                                           float& f0, float& f1, float& f2, float& f3) {
    float Is = S[q], I2 = S[NVOX + q], IJ = S[2 * NVOX + q];
    float Jsum = Js[q], J2 = J2s[q];
    float uI = Is * (1.0f / WINSZ), uJ = Jsum * (1.0f / WINSZ);
    float cross = IJ - Is * uJ;
    float Ivar  = I2 - Is * uI;
    float Jvar  = J2 - Jsum * uJ;
    float den   = Ivar * Jvar + 1e-5f;
    f0 = cross / den;                            // A
    f1 = cross * cross * Jvar / (den * den);     // B
    f2 = f0 * uJ;
    f3 = f1 * uI;
}

// axis-H window sums of the 4 adjoint fields, computed on the fly
__global__ void ws_h_nccbwd(const float* __restrict__ S, const float* __restrict__ Js,
                            const float* __restrict__ J2s, float* __restrict__ out) {
    int l = blockIdx.x * blockDim.x + threadIdx.x;
    if (l >= Ww * Dd) return;
    const long stride = (long)Ww * Dd;
    float s0 = 0.f, s1 = 0.f, s2 = 0.f, s3 = 0.f;
    for (int i = 0; i < Hh + PAD; ++i) {
        if (i < Hh) {
            float f0, f1, f2, f3;
            ncc_fields(S, Js, J2s, l + (long)i * stride, f0, f1, f2, f3);
            s0 += f0; s1 += f1; s2 += f2; s3 += f3;
        }
        int j = i - PAD;
        if (j >= 0) {
            long q = l + (long)j * stride;
            out[q] = s0; out[NVOX + q] = s1; out[2 * NVOX + q] = s2; out[3 * NVOX + q] = s3;
        }
        if (i >= 2 * PAD) {
            float f0, f1, f2, f3;
            ncc_fields(S, Js, J2s, l + (long)(i - 2 * PAD) * stride, f0, f1, f2, f3);
            s0 -= f0; s1 -= f1; s2 -= f2; s3 -= f3;
        }
    }
}

// ---------------- D-axis window pass as banded matmul on WMMA ---------------
// out(16w x 16d) = sum_{r=-1..1} in(16w x 16k @ tile d+r) x Band(16k x 16d)
// One wave32 per output tile; V_WMMA_F32_16X16X4_F32, 4 chained K=4 steps/tile.
__global__ __launch_bounds__(32) void ws_d_wmma(const float* __restrict__ src,
                                                float* __restrict__ dst) {
    int dt = blockIdx.x;              // D tile 0..9
    int wt = blockIdx.y;              // W tile 0..11
    int z  = blockIdx.z;              // c*Hh + h
    int c  = z / Hh, h = z - c * Hh;
    const float* s = src + (long)c * NVOX + (long)h * Ww * Dd;
    float*       o = dst + (long)c * NVOX + (long)h * Ww * Dd;
    int lane = threadIdx.x;
    int row  = lane & 15;             // A: M row / B: N col / D: N col
    int hi   = lane >> 4;             // K half select
    int w0 = wt * 16, d0 = dt * 16;

    v8f acc = {};
    for (int r = -1; r <= 1; ++r) {
        int kt = d0 + 16 * r;
        if (kt < 0 || kt >= Dd) continue;           // zero padding tiles
        const float* arow = s + (long)(w0 + row) * Dd + kt;
        #pragma unroll
        for (int q = 0; q < 4; ++q) {
            int kb = 4 * q + 2 * hi;                // this lane's two K indices
            v2f a; a.x = arow[kb]; a.y = arow[kb + 1];
            // constant banded 0/1 matrix tile: Band[k,n]=1 iff |(d0+n)-(kt+k)|<=4
            int nn = row;
            int g0 = nn - (16 * r + kb);
            int g1 = g0 - 1;
            v2f b;
            b.x = ((g0 < 0 ? -g0 : g0) <= 4) ? 1.0f : 0.0f;
            b.y = ((g1 < 0 ? -g1 : g1) <= 4) ? 1.0f : 0.0f;
            acc = __builtin_amdgcn_wmma_f32_16x16x4_f32(
                false, a, false, b, (short)0, acc, false, false);
        }
    }
    float* ob = o + (long)w0 * Dd + d0;
    #pragma unroll
    for (int rr = 0; rr < 8; ++rr) {
        int M = rr + 8 * hi;
        ob[(long)M * Dd + row] = acc[rr];
    }
}

// ---------------- trilinear sampling helpers --------------------------------
__device__ __forceinline__ float sampX(const float* __restrict__ x, int i, int j, int k) {
    if ((unsigned)i >= (unsigned)Hh || (unsigned)j >= (unsigned)Ww || (unsigned)k >= (unsigned)Dd)
        return 0.f;
    return x[((long)i * Ww + j) * Dd + k];
}

__global__ void warp_fwd(const float* __restrict__ x, const float* __restrict__ flow,
                         float* __restrict__ xw) {
    long p = blockIdx.x * (long)blockDim.x + threadIdx.x;
    if (p >= NVOX) return;
    int d = (int)(p % Dd); long t = p / Dd; int w = (int)(t % Ww); int h = (int)(t / Ww);
    float ch = h + flow[p], cw = w + flow[NVOX + p], cd = d + flow[2 * NVOX + p];
    float fh = floorf(ch), fw = floorf(cw), fd = floorf(cd);
    int i0 = (int)fh, j0 = (int)fw, k0 = (int)fd;
    float ah = ch - fh, aw = cw - fw, ad = cd - fd;
    float X000 = sampX(x, i0, j0, k0),     X001 = sampX(x, i0, j0, k0 + 1);
    float X010 = sampX(x, i0, j0 + 1, k0), X011 = sampX(x, i0, j0 + 1, k0 + 1);
    float X100 = sampX(x, i0 + 1, j0, k0),     X101 = sampX(x, i0 + 1, j0, k0 + 1);
    float X110 = sampX(x, i0 + 1, j0 + 1, k0), X111 = sampX(x, i0 + 1, j0 + 1, k0 + 1);
    float v0 = (X000 * (1 - ad) + X001 * ad) * (1 - aw) + (X010 * (1 - ad) + X011 * ad) * aw;
    float v1 = (X100 * (1 - ad) + X101 * ad) * (1 - aw) + (X110 * (1 - ad) + X111 * ad) * aw;
    xw[p] = v0 * (1 - ah) + v1 * ah;
}

// -------- fused: dI combine + warp backward + regularizer gradient ----------
// dL/dI_q = -(2/N)[ J_q*WS(A) - WS(A uJ) - I_q*WS(B) + WS(B uI) ]
__global__ void flow_grad(const float* __restrict__ x, const float* __restrict__ flow,
                          const float* __restrict__ WS, const float* __restrict__ y,
                          const float* __restrict__ xw, float* __restrict__ g) {
    long p = blockIdx.x * (long)blockDim.x + threadIdx.x;
    if (p >= NVOX) return;
    int d = (int)(p % Dd); long t = p / Dd; int w = (int)(t % Ww); int h = (int)(t / Ww);

    float gdot = y[p] * WS[p] - WS[2 * NVOX + p] - xw[p] * WS[NVOX + p] + WS[3 * NVOX + p];
    float gI = (-2.0f / (float)NVOX) * gdot;

    float ch = h + flow[p], cw = w + flow[NVOX + p], cd = d + flow[2 * NVOX + p];
    float fh = floorf(ch), fw = floorf(cw), fd = floorf(cd);
    int i0 = (int)fh, j0 = (int)fw, k0 = (int)fd;
    float ah = ch - fh, aw = cw - fw, ad = cd - fd;
    float X000 = sampX(x, i0, j0, k0),     X001 = sampX(x, i0, j0, k0 + 1);
    float X010 = sampX(x, i0, j0 + 1, k0), X011 = sampX(x, i0, j0 + 1, k0 + 1);
    float X100 = sampX(x, i0 + 1, j0, k0),     X101 = sampX(x, i0 + 1, j0, k0 + 1);
    float X110 = sampX(x, i0 + 1, j0 + 1, k0), X111 = sampX(x, i0 + 1, j0 + 1, k0 + 1);
    float dvdh = (X100 - X000) * (1 - aw) * (1 - ad) + (X110 - X010) * aw * (1 - ad)
               + (X101 - X001) * (1 - aw) * ad       + (X111 - X011) * aw * ad;
    float dvdw = ((X010 - X000) * (1 - ad) + (X011 - X001) * ad) * (1 - ah)
               + ((X110 - X100) * (1 - ad) + (X111 - X101) * ad) * ah;
    float dvdd = ((X001 - X000) * (1 - aw) + (X011 - X010) * aw) * (1 - ah)
               + ((X101 - X100) * (1 - aw) + (X111 - X110) * aw) * ah;
    // grad3d_l2 gradient: mean over (3,Hdim-1,W,D) etc., averaged over 3 axes
    const float sH = 2.0f / (3.0f * (float)((long)(Hh - 1) * Ww * Dd));
    const float sW = 2.0f / (3.0f * (float)((long)Hh * (Ww - 1) * Dd));
    const float sD = 2.0f / (3.0f * (float)((long)Hh * Ww * (Dd - 1)));
    #pragma unroll
    for (int c = 0; c < 3; ++c) {
        long q = (long)c * NVOX + p;
        float f = flow[q];
        float r = 0.f;
        if (h > 0)      r += sH * (f - flow[q - (long)Ww * Dd]);
        if (h < Hh - 1) r -= sH * (flow[q + (long)Ww * Dd] - f);
        if (w > 0)      r += sW * (f - flow[q - Dd]);
        if (w < Ww - 1) r -= sW * (flow[q + Dd] - f);
        if (d > 0)      r += sD * (f - flow[q - 1]);
        if (d < Dd - 1) r -= sD * (flow[q + 1] - f);
        float gw = (c == 0) ? dvdh : (c == 1) ? dvdw : dvdd;
        g[q] = gI * gw + r;
    }
}

// ---------------- Adam(amsgrad) update --------------------------------------
__global__ void adam_up(const float* __restrict__ g, float* __restrict__ flow,
                        float* __restrict__ m, float* __restrict__ v, float* __restrict__ vh,
                        float b1corr /*1/(1-b1^t)*/, float s2 /*sqrt(1-b2^t)*/) {
    long i = blockIdx.x * (long)blockDim.x + threadIdx.x;
    if (i >= 3 * NVOX) return;
    float gi = g[i];
    float mi = B1c * m[i] + (1.f - B1c) * gi;
    float vi = B2c * v[i] + (1.f - B2c) * gi * gi;
    float vhi = fmaxf(vh[i], vi);
    m[i] = mi; v[i] = vi; vh[i] = vhi;
    float den = sqrtf(vhi) / s2 + EPSc;
    flow[i] -= LRc * (mi * b1corr) / den;
}

// ---------------- deterministic two-stage MSE reduction ---------------------
__global__ void mse_part(const float* __restrict__ flow, const float* __restrict__ initf,
                         float* __restrict__ part) {
    __shared__ float red[256];
    float acc = 0.f;
    for (long i = blockIdx.x * 256L + threadIdx.x; i < 3 * NVOX; i += 256L * gridDim.x) {
        float dl = flow[i] - initf[i];
        acc += dl * dl;
    }
    red[threadIdx.x] = acc; __syncthreads();
    for (int s = 128; s > 0; s >>= 1) {
        if (threadIdx.x < s) red[threadIdx.x] += red[threadIdx.x + s];
        __syncthreads();
    }
    if (threadIdx.x == 0) part[blockIdx.x] = red[0];
}

__global__ void mse_final(const float* __restrict__ part, int nblk, float* __restrict__ out) {
    __shared__ float red[256];
    float acc = 0.f;
    for (int i = threadIdx.x; i < nblk; i += 256) acc += part[i];
    red[threadIdx.x] = acc; __syncthreads();
    for (int s = 128; s > 0; s >>= 1) {
        if (threadIdx.x < s) red[threadIdx.x] += red[threadIdx.x + s];
        __syncthreads();
    }
    if (threadIdx.x == 0) out[0] = red[0] / (float)(3 * NVOX);
}

// ---------------- host orchestration ----------------------------------------
extern "C" void kernel_launch(void* const* d_in, const int* in_sizes, int n_in,
                              void* d_out, int out_size, void* d_ws, size_t ws_size,
                              hipStream_t stream) {
    (void)in_sizes; (void)n_in; (void)out_size; (void)ws_size;
    const float* x    = (const float*)d_in[0];
    const float* y    = (const float*)d_in[1];
    const float* init = (const float*)d_in[2];
    float* ws = (float*)d_ws;

    float* flow = ws;                   // 3N
    float* m    = ws + 3 * NVOX;        // 3N
    float* v    = ws + 6 * NVOX;        // 3N
    float* vh   = ws + 9 * NVOX;        // 3N
    float* Jsum = ws + 12 * NVOX;       // 1N   (J2sum at 13N, contiguous 2-ch)
    float* J2s  = ws + 13 * NVOX;       // 1N
    float* xw   = ws + 14 * NVOX;       // 1N
    float* bufA = ws + 15 * NVOX;       // 4N
    float* bufB = ws + 19 * NVOX;       // 4N   (also reused as grad / partials)

    const int T = 256;
    const long gN  = (NVOX + T - 1) / T;
    const long g3N = (3 * NVOX + T - 1) / T;
    const int gl0 = (Ww * Dd + T - 1) / T;   // axis-H lines
    const int gl1 = (Hh * Dd + T - 1) / T;   // axis-W lines
    dim3 wm2(Dd / 16, Ww / 16, 2 * Hh);
    dim3 wm3(Dd / 16, Ww / 16, 3 * Hh);
    dim3 wm4(Dd / 16, Ww / 16, 4 * Hh);

    init_state<<<g3N, T, 0, stream>>>(init, flow, m, v, vh);

    // loop-invariant J moments: fused [y, y^2] axis-H pass, then W, then D(WMMA)
    ws_h_pointJ<<<gl0, T, 0, stream>>>(y, bufB);
    ws_line_w<<<dim3(gl1, 2), T, 0, stream>>>(bufB, bufA);
    ws_d_wmma<<<wm2, 32, 0, stream>>>(bufA, Jsum);   // writes Jsum, J2s

    for (int t = 1; t <= 5; ++t) {
        warp_fwd<<<gN, T, 0, stream>>>(x, flow, xw);
        // forward moments: fused [I, I^2, I*J] axis-H, then W, then D(WMMA)
        ws_h_point3<<<gl0, T, 0, stream>>>(xw, y, bufB);
        ws_line_w<<<dim3(gl1, 3), T, 0, stream>>>(bufB, bufA);
        ws_d_wmma<<<wm3, 32, 0, stream>>>(bufA, bufB);             // I_sum,I2_sum,IJ_sum
        // backward: fused NCC-adjoint fields + axis-H, then W, then D(WMMA)
        ws_h_nccbwd<<<gl0, T, 0, stream>>>(bufB, Jsum, J2s, bufA);
        ws_line_w<<<dim3(gl1, 4), T, 0, stream>>>(bufA, bufB);
        ws_d_wmma<<<wm4, 32, 0, stream>>>(bufB, bufA);             // WS of 4 fields -> bufA
        // fused dI combine + warp-backward + reg grad
        flow_grad<<<gN, T, 0, stream>>>(x, flow, bufA, y, xw, bufB);
        float b1corr = 1.0f / (1.0f - powf(B1c, (float)t));
        float s2     = sqrtf(1.0f - powf(B2c, (float)t));
        adam_up<<<g3N, T, 0, stream>>>(bufB, flow, m, v, vh, b1corr, s2);
    }

    mse_part<<<1024, 256, 0, stream>>>(flow, init, bufB);
    mse_final<<<1, 256, 0, stream>>>(bufB, 1024, (float*)d_out);
}